// PoincareEmbedding_21165598834714
// MI455X (gfx1250) — compile-verified
//
#include <hip/hip_runtime.h>
#include <math.h>

typedef __attribute__((ext_vector_type(2))) float v2f;
typedef __attribute__((ext_vector_type(8))) float v8f;

#define PB_EPS     1e-7f
#define PB_MIN_DEN 1e-15f

// matrix: [N, 32] f32, items: [B, L] i32, origin: [B] i32, out: [B, L] f32
__global__ __launch_bounds__(128) void poincare_dist_kernel(
    const float* __restrict__ matrix,
    const int*   __restrict__ items,
    const int*   __restrict__ origin,
    float*       __restrict__ out,
    int L, int nTiles)
{
    __shared__ float sy[32];
    __shared__ float sy2;

    const int b    = blockIdx.x;
    const int tid  = threadIdx.x;
    const int lane = tid & 31;
    const int wave = tid >> 5;
    const int nWaves = blockDim.x >> 5;

    // Stage origin row y and y2 = y.y (wave 0 only), then share via LDS.
    if (wave == 0) {
        const float yd = matrix[(size_t)origin[b] * 32 + lane];
        sy[lane] = yd;
        float p = yd * yd;
        #pragma unroll
        for (int off = 16; off > 0; off >>= 1) p += __shfl_xor(p, off, 32);
        if (lane == 0) sy2 = p;
    }
    __syncthreads();
    const float y2 = sy2;

    // WMMA lane roles: lane = m + 16*h. Lane (m,h) supplies A[m][K=2h..2h+1]
    // of each 16x4 K-slab, i.e. float2 at row float-offset 4s+2h.
    const int h = lane >> 4;       // half-wave (K sub-block)
    const int m = lane & 15;       // candidate row fed into A
    const int r = lane & 7;        // accumulator VGPR index this lane extracts
    const bool owner = ((lane >> 3) & 1) == h;  // lanes 0-7 and 24-31
    const int  cown  = r + 8 * h;               // candidate this lane finalizes

    for (int t = wave; t < nTiles; t += nWaves) {
        const int l0 = t * 16;
        int l = l0 + m;
        if (l >= L) l = L - 1;                       // clamp tail gathers (dup rows)
        const int idx = items[(size_t)b * L + l];
        const float* row = matrix + (size_t)idx * 32 + 2 * h;

        v8f accY = {};   // D[i][j] = x_i . y   (y broadcast to all 16 columns)
        v8f accG = {};   // D[i][j] = x_i . x_j (Gram; diagonal = ||x_i||^2)
        #pragma unroll
        for (int s = 0; s < 8; ++s) {
            // A operand: lane (m,h) holds X[m][4s+2h], X[m][4s+2h+1]
            const v2f a = *(const v2f*)(row + 4 * s);
            // B operand for x.y: every column = y  -> lane-uniform per half-wave
            v2f bv;
            bv.x = sy[4 * s + 2 * h];
            bv.y = sy[4 * s + 2 * h + 1];
            accY = __builtin_amdgcn_wmma_f32_16x16x4_f32(
                false, a, false, bv, (short)0, accY, false, false);
            // A register pair doubles as B layout of X^T -> Gram matrix
            accG = __builtin_amdgcn_wmma_f32_16x16x4_f32(
                false, a, false, a, (short)0, accG, false, false);
        }

        // Epilogue: only 16 owner lanes, strictly after all WMMAs (EXEC all-1s
        // during matrix ops as the ISA requires).
        if (owner && (l0 + cown) < L) {
            float xy = accY[0];
            float a2 = accG[0];
            #pragma unroll
            for (int i = 1; i < 8; ++i) {
                if (r == i) { xy = accY[i]; a2 = accG[i]; }
            }
            // geoopt mobius_add(-x, y) distance, C = 1:
            //   ay    = <-x, y>
            //   num   = (1 + 2*ay + y2)*(-x) + (1 - a2)*y
            //   den   = 1 + 2*ay + a2*y2
            //   ||num||^2 = al^2*a2 + 2*al*be*ay + be^2*y2   (closed form)
            const float ay = -xy;
            const float al = 1.0f + 2.0f * ay + y2;
            const float be = 1.0f - a2;
            float den = 1.0f + 2.0f * ay + a2 * y2;
            den = fmaxf(den, PB_MIN_DEN);
            float nn = al * al * a2 + 2.0f * al * be * ay + be * be * y2;
            nn = fmaxf(nn, 0.0f);
            float z = sqrtf(nn) / den;              // = sqrt(sum m^2)
            z = fminf(z, 1.0f - PB_EPS);            // z >= 0, only upper clip binds
            out[(size_t)b * L + l0 + cown] = -2.0f * atanhf(z);
        }
    }
}

extern "C" void kernel_launch(void* const* d_in, const int* in_sizes, int n_in,
                              void* d_out, int out_size, void* d_ws, size_t ws_size,
                              hipStream_t stream) {
    (void)n_in; (void)d_ws; (void)ws_size; (void)out_size;
    const float* matrix = (const float*)d_in[0];   // [N, 32] f32
    const int*   items  = (const int*)d_in[1];     // [B, L] i32
    const int*   origin = (const int*)d_in[2];     // [B]    i32
    float*       out    = (float*)d_out;           // [B, L] f32

    const int B = in_sizes[2];
    const int L = in_sizes[1] / B;
    const int nTiles = (L + 15) / 16;

    poincare_dist_kernel<<<B, 128, 0, stream>>>(matrix, items, origin, out, L, nTiles);
}